// LIF_hh_neuron_30013231464739
// MI455X (gfx1250) — compile-verified
//
#include <hip/hip_runtime.h>

// ---- problem constants (from reference setup_inputs) ----
constexpr int B_  = 256;
constexpr int T_  = 15;
constexpr int IN_ = 4096;
constexpr int C_  = 4096;

// ---- tiling ----
constexpr int BM  = 32;      // b rows per workgroup
constexpr int BN  = 64;      // c cols per workgroup
constexpr int KB  = 128;     // K block
constexpr int LDW = KB + 4;  // LDS row stride (bank step 4 -> conflict-free frags)

// byte-step constants
constexpr unsigned RSTEP  = 8u * LDW * 4u;            // 8 LDS rows    = 4224 B
constexpr unsigned XBUF   = (unsigned)(BM * LDW * 4); // one X buffer  = 16896 B
constexpr unsigned WSBUF  = (unsigned)(BN * LDW * 4); // one W buffer  = 33792 B
constexpr unsigned WGSTEP = 8u * IN_ * 4u;            // 8 W rows in global
constexpr unsigned XGSTEP = 8u * T_ * IN_ * 4u;       // 8 x rows in global
constexpr unsigned TSTEP  = (unsigned)(IN_ * 4);      // one timestep in global

#define THRESH 0.8f
#define DECAY  0.2f

typedef float v2f __attribute__((ext_vector_type(2)));
typedef float v8f __attribute__((ext_vector_type(8)));

// Async DMA: memory -> LDS, 16B per lane, no VGPR data path. Tracked by ASYNCcnt.
#define ASYNC_LOAD_B128(ldsoff_u32, goff_u32, base_ptr)                         \
    asm volatile("global_load_async_to_lds_b128 %0, %1, %2"                     \
                 :: "v"(ldsoff_u32), "v"(goff_u32), "s"(base_ptr) : "memory")

// Wait until at most `n` async transfers outstanding (completions are in-order).
#define WAIT_ASYNC(n) asm volatile("s_wait_asynccnt " #n ::: "memory")

__global__ __launch_bounds__(256, 1)
void lif_fused_kernel(const float* __restrict__ x,     // [B, T, IN]
                      const float* __restrict__ Wfc,   // [C, IN]
                      const float* __restrict__ bfc,   // [C]
                      const float* __restrict__ Wlif,  // [1,3]
                      const float* __restrict__ blif,  // [1]
                      float* __restrict__ out)         // [B, T, C] spikes
{
    __shared__ float Ws[2][BN][LDW];     // W tile, double buffered over K blocks
    __shared__ float Xs[4][BM][LDW];     // X tile, quad buffered over t

    const int tid  = threadIdx.x;
    const int lane = tid & 31;
    const int wave = tid >> 5;
    const int lo   = lane & 15;
    const int hi   = lane >> 4;
    const int br   = wave & 1;           // 16-row b sub-tile
    const int cc   = wave >> 1;          // 16-col c sub-tile

    const int c0 = blockIdx.x * BN;
    const int b0 = blockIdx.y * BM;

    const int cCh      = c0 + cc * 16 + lo;
    const int bRowBase = b0 + br * 16;

    // Staging geometry: thread owns 16B column `scol` of rows {wave + 8j}.
    const int scol = lane << 2;

    // Base byte offsets; all j / buffer / t steps are compile-time constants.
    const unsigned wgo0 = (unsigned)(((c0 + wave) * IN_ + scol) * 4);
    const unsigned xgo0 = (unsigned)(((b0 + wave) * (T_ * IN_) + scol) * 4);
    const unsigned wlo0 = (unsigned)(unsigned long long)&Ws[0][wave][scol];
    const unsigned xlo0 = (unsigned)(unsigned long long)&Xs[0][wave][scol];

    // DMA helpers -------------------------------------------------------------
    auto dmaW = [&](int buf, unsigned kb) {
        #pragma unroll
        for (int j = 0; j < 8; ++j)
            ASYNC_LOAD_B128(wlo0 + (unsigned)buf * WSBUF + (unsigned)j * RSTEP,
                            wgo0 + (unsigned)j * WGSTEP + kb, Wfc);
    };
    auto dmaX = [&](int buf, int t, unsigned kb) {
        #pragma unroll
        for (int j = 0; j < 4; ++j)
            ASYNC_LOAD_B128(xlo0 + (unsigned)buf * XBUF + (unsigned)j * RSTEP,
                            xgo0 + (unsigned)j * XGSTEP + (unsigned)t * TSTEP + kb, x);
    };

    const v8f zero8 = {0.f,0.f,0.f,0.f,0.f,0.f,0.f,0.f};
    v8f acc[T_];
    #pragma unroll
    for (int t = 0; t < T_; ++t) acc[t] = zero8;

    // ---- initial DMAs: W(block 0) -> Ws[0], X(0) -> buf0, X(1) -> buf1 ----
    dmaW(0, 0u);
    dmaX(0, 0, 0u);
    dmaX(1, 1, 0u);

    for (int k0 = 0, wb = 0; k0 < IN_; k0 += KB, wb ^= 1) {
        const unsigned k0b = (unsigned)(k0 * 4);
        WAIT_ASYNC(0);
        __syncthreads();  // Ws[wb], X(0), X(1) visible to all waves

        // ---- 7 windows of two timesteps each: two independent WMMA chains ----
        #pragma unroll
        for (int p = 0; p < 7; ++p) {
            const int t0 = 2 * p, t1 = 2 * p + 1;

            if (p < 6) {  // prefetch the next pair (issued FIRST -> retired first)
                dmaX((t0 + 2) & 3, t0 + 2, k0b);
                dmaX((t1 + 2) & 3, t1 + 2, k0b);
            } else {      // prefetch the tail timestep
                dmaX(14 & 3, 14, k0b);
            }
            if (p == 0 && k0 + KB < IN_)
                dmaW(wb ^ 1, k0b + KB * 4);   // next block's W; retires by p=1's wait

            const float* a0r = &Xs[t0 & 3][br * 16 + lo][2 * hi];
            const float* a1r = &Xs[t1 & 3][br * 16 + lo][2 * hi];
            const float* wr  = &Ws[wb][cc * 16 + lo][2 * hi];
            #pragma unroll
            for (int s = 0; s < KB / 4; ++s) {
                v2f bb = *(const v2f*)(wr  + s * 4);   // B fragment shared by both chains
                v2f a0 = *(const v2f*)(a0r + s * 4);
                v2f a1 = *(const v2f*)(a1r + s * 4);
                acc[t0] = __builtin_amdgcn_wmma_f32_16x16x4_f32(
                    false, a0, false, bb, (short)0, acc[t0], false, false);
                acc[t1] = __builtin_amdgcn_wmma_f32_16x16x4_f32(
                    false, a1, false, bb, (short)0, acc[t1], false, false);
            }

            if (p == 0 && k0 + KB < IN_) {
                WAIT_ASYNC(8);   // drain the 8 X transfers; W's 8 keep flying
            } else {
                WAIT_ASYNC(0);
            }
            __syncthreads();
        }

        // ---- tail window: t = 14 (buffer 2); prefetch next block's X(0), X(1) ----
        if (k0 + KB < IN_) {
            dmaX(0, 0, k0b + KB * 4);
            dmaX(1, 1, k0b + KB * 4);
        }
        {
            const float* a0r = &Xs[14 & 3][br * 16 + lo][2 * hi];
            const float* wr  = &Ws[wb][cc * 16 + lo][2 * hi];
            #pragma unroll
            for (int s = 0; s < KB / 4; ++s) {
                v2f bb = *(const v2f*)(wr  + s * 4);
                v2f a0 = *(const v2f*)(a0r + s * 4);
                acc[14] = __builtin_amdgcn_wmma_f32_16x16x4_f32(
                    false, a0, false, bb, (short)0, acc[14], false, false);
            }
        }
        // next-block X DMAs are waited + published at the top of the next k0 iter
    }

    // ---- LIF recurrence, fully in registers ----
    const float w0 = Wlif[0], w1 = Wlif[1], w2 = Wlif[2];
    const float bl = blif[0];
    const float bias = bfc[cCh];

    float m0[8], m1[8], m2[8], m3[8], sp[8];
    #pragma unroll
    for (int r = 0; r < 8; ++r) { m0[r] = m1[r] = m2[r] = m3[r] = sp[r] = 0.f; }

    #pragma unroll
    for (int t = 0; t < T_; ++t) {
        #pragma unroll
        for (int r = 0; r < 8; ++r) {
            float xt   = acc[t][r] + bias;                           // fc_out[b,t,c]
            float x4   = w0 * m0[r] + w1 * m1[r] + w2 * m2[r] + bl;  // from OLD mem
            float keep = DECAY * (1.f - sp[r]);
            float n0 = m0[r] * keep + xt;
            float n1 = m1[r] * keep + xt;
            float n2 = m2[r] * keep + xt;
            float n3 = m3[r] * keep + x4;
            float mem1 = w0 * n0 + w1 * n1 + w2 * n2 + bl + n3;
            float s = (mem1 > THRESH) ? 1.f : 0.f;
            m0[r] = n0; m1[r] = n1; m2[r] = n2; m3[r] = n3; sp[r] = s;

            int bRow = bRowBase + r + 8 * hi;   // C/D layout: VGPR r <-> M = r + 8*hi
            out[(size_t)bRow * (T_ * C_) + (size_t)t * C_ + cCh] = s;
        }
    }
}

extern "C" void kernel_launch(void* const* d_in, const int* in_sizes, int n_in,
                              void* d_out, int out_size, void* d_ws, size_t ws_size,
                              hipStream_t stream) {
    const float* x    = (const float*)d_in[0];
    const float* Wfc  = (const float*)d_in[1];
    const float* bfc  = (const float*)d_in[2];
    const float* Wlif = (const float*)d_in[3];
    const float* blif = (const float*)d_in[4];
    float* out = (float*)d_out;

    dim3 grid(C_ / BN, B_ / BM);   // 64 x 8 = 512 workgroups
    dim3 block(256);               // 8 wave32s
    lif_fused_kernel<<<grid, block, 0, stream>>>(x, Wfc, bfc, Wlif, blif, out);
}